// FusionIntegrator_28613072126542
// MI455X (gfx1250) — compile-verified
//
#include <hip/hip_runtime.h>

// FusionIntegrator for MI455X (gfx1250).
// Phase 1: zero accumulators (reuse d_out regions, graph-replay safe).
// Phase 2: scatter-add 8-float feature rows + counts into 256^3 grid with
//          no-return global_atomic_add_f32.
// Phase 3: per-voxel mean -> L2 normalize -> running-mean blend, streamed
//          with 128-bit non-temporal loads/stores (don't thrash L2).

typedef float v4f __attribute__((ext_vector_type(4)));

namespace {
constexpr int       kXS   = 256;
constexpr long long kNVox = 256LL * 256LL * 256LL;   // 16,777,216
constexpr int       kNF   = 8;
constexpr int       kNS   = 4;                        // N_SAMP
constexpr int       kS    = 2 * kNS + 1;              // 9 samples per ray
constexpr int       kP    = 76800;                    // points
constexpr int       kN    = kS * kP;                  // 691,200 scatter rows
constexpr float     kEPS  = 1e-12f;
}

__global__ void fi_zero(v4f* __restrict__ acc, int n4) {
  int i = blockIdx.x * blockDim.x + threadIdx.x;
  if (i < n4) {
    v4f z = (v4f)(0.0f);
    __builtin_nontemporal_store(z, acc + i);
  }
}

__global__ void fi_scatter(const float* __restrict__ updates,   // (NF, P, S) flat
                           const float* __restrict__ vpoints,   // (P, 3)
                           const float* __restrict__ veye,      // (P, 3)
                           const int*   __restrict__ mask,      // (P,)
                           float* __restrict__ featAcc,         // (nvox, NF)
                           float* __restrict__ cntAcc) {        // (nvox,)
  int n = blockIdx.x * blockDim.x + threadIdx.x;
  if (n >= kN) return;

  // Reference row ordering quirk: positions/mask use (s,p) order, features
  // use (p,s) order. For scatter row n: p = n % P, s = n / P, and the
  // feature row is simply updates[f*N + n] (fully coalesced).
  int p = n % kP;
  int s = n / kP;
  if (mask[p] == 0) return;

  float vx = vpoints[3 * p + 0];
  float vy = vpoints[3 * p + 1];
  float vz = vpoints[3 * p + 2];
  float dx = vx - veye[3 * p + 0];
  float dy = vy - veye[3 * p + 1];
  float dz = vz - veye[3 * p + 2];
  float inv = 1.0f / fmaxf(sqrtf(dx * dx + dy * dy + dz * dz), kEPS);
  float off = (float)(s - kNS) * inv;
  float px = fmaf(off, dx, vx);
  float py = fmaf(off, dy, vy);
  float pz = fmaf(off, dz, vz);

  int bx = (int)floorf(px);
  int by = (int)floorf(py);
  int bz = (int)floorf(pz);

  float u[kNF];
#pragma unroll
  for (int f = 0; f < kNF; ++f)
    u[f] = updates[(long long)f * kN + n];

#pragma unroll
  for (int c = 0; c < 8; ++c) {
    int x = bx + ((c >> 2) & 1);
    int y = by + ((c >> 1) & 1);
    int z = bz + (c & 1);
    if ((unsigned)x < (unsigned)kXS && (unsigned)y < (unsigned)kXS &&
        (unsigned)z < (unsigned)kXS) {
      long long lin = ((long long)x << 16) | (y << 8) | z;  // (x*256+y)*256+z
      float* fb = featAcc + lin * kNF;
#pragma unroll
      for (int f = 0; f < kNF; ++f)
        atomicAdd(fb + f, u[f]);          // no-return global_atomic_add_f32
      atomicAdd(cntAcc + lin, 1.0f);
    }
  }
}

__global__ void fi_finalize(const float* __restrict__ grid,     // (nvox, NF)
                            const float* __restrict__ count,    // (nvox,)
                            float* __restrict__ vol,            // in: featAcc, out: new_vol
                            float* __restrict__ touched,        // in: cntAcc,  out: touched
                            float* __restrict__ newCount) {     // out
  long long v = (long long)blockIdx.x * blockDim.x + threadIdx.x;
  if (v >= kNVox) return;

  float cnt = touched[v];
  const v4f* gp = (const v4f*)(grid + v * kNF);
  v4f g0 = __builtin_nontemporal_load(gp);
  v4f g1 = __builtin_nontemporal_load(gp + 1);
  float c = count[v];

  v4f o0 = g0, o1 = g1;
  float t = 0.0f, nc = c;

  if (cnt > 0.0f) {
    const v4f* fp = (const v4f*)(vol + v * kNF);
    v4f f0 = fp[0];
    v4f f1 = fp[1];
    float ic = 1.0f / cnt;
    float a0 = f0.x * ic, a1 = f0.y * ic, a2 = f0.z * ic, a3 = f0.w * ic;
    float a4 = f1.x * ic, a5 = f1.y * ic, a6 = f1.z * ic, a7 = f1.w * ic;
    float nrm = sqrtf(a0 * a0 + a1 * a1 + a2 * a2 + a3 * a3 +
                      a4 * a4 + a5 * a5 + a6 * a6 + a7 * a7);
    float innrm = 1.0f / fmaxf(nrm, kEPS);
    float iw = 1.0f / (c + 1.0f);
    o0.x = (c * g0.x + a0 * innrm) * iw;
    o0.y = (c * g0.y + a1 * innrm) * iw;
    o0.z = (c * g0.z + a2 * innrm) * iw;
    o0.w = (c * g0.w + a3 * innrm) * iw;
    o1.x = (c * g1.x + a4 * innrm) * iw;
    o1.y = (c * g1.y + a5 * innrm) * iw;
    o1.z = (c * g1.z + a6 * innrm) * iw;
    o1.w = (c * g1.w + a7 * innrm) * iw;
    t = 1.0f;
    nc = c + 1.0f;
  }

  v4f* op = (v4f*)(vol + v * kNF);
  __builtin_nontemporal_store(o0, op);
  __builtin_nontemporal_store(o1, op + 1);
  touched[v]  = t;
  newCount[v] = nc;
}

extern "C" void kernel_launch(void* const* d_in, const int* in_sizes, int n_in,
                              void* d_out, int out_size, void* d_ws, size_t ws_size,
                              hipStream_t stream) {
  (void)in_sizes; (void)n_in; (void)out_size; (void)d_ws; (void)ws_size;

  const float* updates = (const float*)d_in[0];  // (1, NF, P, S)
  const float* vpoints = (const float*)d_in[1];  // (1, P, 3)
  const float* veye    = (const float*)d_in[2];  // (1, P, 3)
  const int*   mask    = (const int*)d_in[3];    // (1, P) bool -> int
  const float* grid    = (const float*)d_in[4];  // (256,256,256,NF)
  const float* count   = (const float*)d_in[5];  // (256,256,256)

  float* out      = (float*)d_out;
  float* vol      = out;                                // nvox*NF (accumulator, then new_vol)
  float* touchedp = out + kNVox * kNF;                  // nvox    (count accum, then touched)
  float* newCount = out + kNVox * kNF + kNVox;          // nvox

  // Zero the feature + count accumulator regions: nvox*9 floats.
  const int n4 = (int)(kNVox * 9 / 4);                  // 37,748,736 float4s
  fi_zero<<<(n4 + 255) / 256, 256, 0, stream>>>((v4f*)out, n4);

  fi_scatter<<<(kN + 255) / 256, 256, 0, stream>>>(updates, vpoints, veye, mask,
                                                   vol, touchedp);

  fi_finalize<<<(int)(kNVox / 256), 256, 0, stream>>>(grid, count, vol,
                                                      touchedp, newCount);
}